// GraphConvolution_14903536517267
// MI455X (gfx1250) — compile-verified
//
#include <hip/hip_runtime.h>
#include <cstdint>

// ---------------------------------------------------------------------------
// GCN layer: out = A_coo @ (X @ W) + bias
//   X: [100000, 256] f32   W: [256,128] f32   edges: 1.6M COO  out: [100000,128]
// Stage 1 (HBM-bound, ~154MB):  support = X@W via V_WMMA_F32_16X16X4_F32
//   - W staged in LDS in fragment-native [k2][col][2] layout (+32-float pad
//     per row) so B-fragments are single conflict-free ds_load_b64s.
//   - X streamed with non-temporal loads to keep support resident in L2.
// Stage 2 (L2-resident):  bias init + per-edge gather/scale/atomic-scatter
//   (support 51.2MB + out 51.2MB both fit in 192MB L2).
// ---------------------------------------------------------------------------

typedef float v2f __attribute__((ext_vector_type(2)));
typedef float v8f __attribute__((ext_vector_type(8)));

#define N_NODES  100000
#define N_EDGES  1600000
#define IN_FEAT  256
#define OUT_FEAT 128

// LDS row stride per k2 (pair of K values): 128 cols * 2 floats + 32 pad floats.
// The pad shifts laneHi=1 reads by exactly 32 banks -> conflict-free b64 loads.
#define LDS_STRIDE 288

// ---------------------------------------------------------------------------
// Stage 1: support = X @ W   (fp32 WMMA 16x16x4)
// Block = 256 threads = 8 waves; each wave computes a 16-row x 128-col strip.
// ---------------------------------------------------------------------------
__global__ __launch_bounds__(256) void gcn_gemm_wmma(
    const float* __restrict__ X, const float* __restrict__ W,
    float* __restrict__ S)
{
  __shared__ float ldsW[(IN_FEAT / 2) * LDS_STRIDE];   // 128*288*4 = 144 KB

  const int tid = threadIdx.x;
  // Stage W into fragment-native layout: ldsW[k2*288 + col*2 + j] = W[2*k2+j][col].
  // 16384 pairs / 256 threads = 64 float2 stores per thread, conflict-free.
  #pragma unroll
  for (int i = 0; i < (IN_FEAT / 2 * OUT_FEAT) / 256; ++i) {
    const int p   = i * 256 + tid;
    const int k2  = p >> 7;          // 0..127
    const int col = p & 127;
    v2f w;
    w.x = W[(2 * k2    ) * OUT_FEAT + col];
    w.y = W[(2 * k2 + 1) * OUT_FEAT + col];
    *(v2f*)&ldsW[k2 * LDS_STRIDE + col * 2] = w;
  }
  __syncthreads();

  const int wave   = tid >> 5;
  const int lane   = tid & 31;
  const int laneLo = lane & 15;
  const int laneHi = lane >> 4;      // 0: K%4 in {0,1}, M 0-7 ; 1: K%4 in {2,3}, M 8-15

  const int rowBase = blockIdx.x * 128 + wave * 16;
  int arow = rowBase + laneLo;
  if (arow >= N_NODES) arow = N_NODES - 1;   // clamp (WMMA requires EXEC all-ones)

  v8f acc[8];
  #pragma unroll
  for (int t = 0; t < 8; ++t) acc[t] = v8f{};

  const float* __restrict__ xrow = X + (size_t)arow * IN_FEAT;

  for (int s = 0; s < IN_FEAT / 4; ++s) {
    // A fragment 16x4: lane holds K = 4s + laneHi*2 + {0,1} of row arow.
    // Non-temporal: X is read exactly once; don't evict support from L2.
    const v2f a = __builtin_nontemporal_load((const v2f*)&xrow[4 * s + laneHi * 2]);

    // B fragments: one aligned ds_load_b64 each, straight into a VGPR pair.
    const float* __restrict__ bbase = &ldsW[(2 * s + laneHi) * LDS_STRIDE + laneLo * 2];
    #pragma unroll
    for (int t = 0; t < 8; ++t) {
      const v2f b = *(const v2f*)&bbase[t * 32];
      acc[t] = __builtin_amdgcn_wmma_f32_16x16x4_f32(
          /*neg_a=*/false, a, /*neg_b=*/false, b,
          /*c_mod=*/0, acc[t], /*reuse_a=*/false, /*reuse_b=*/false);
    }
  }

  // D layout: VGPR i -> M = i (lanes 0-15) / M = 8+i (lanes 16-31); N = laneLo.
  #pragma unroll
  for (int t = 0; t < 8; ++t) {
    const int col = t * 16 + laneLo;
    #pragma unroll
    for (int i = 0; i < 8; ++i) {
      const int r = rowBase + laneHi * 8 + i;
      if (r < N_NODES)
        S[(size_t)r * OUT_FEAT + col] = acc[t][i];
    }
  }
}

// ---------------------------------------------------------------------------
// Stage 2a: out[n, c] = bias[c]   (12.8M elements, one per thread, exact grid)
// ---------------------------------------------------------------------------
__global__ __launch_bounds__(256) void gcn_bias_init(
    const float* __restrict__ bias, float* __restrict__ out)
{
  const int i = blockIdx.x * 256 + threadIdx.x;
  out[i] = bias[i & (OUT_FEAT - 1)];
}

// ---------------------------------------------------------------------------
// Stage 2b: per-edge gather/scale/atomic-scatter. One wave per edge; each
// lane owns a float4 of the 128-wide row. support & out are L2-resident,
// so gathers + fp32 atomics stay on-chip.
// ---------------------------------------------------------------------------
__device__ __forceinline__ void global_fadd(float* p, float v) {
  // No-return fp32 hardware atomic at device scope (not a CAS loop).
  asm volatile("global_atomic_add_f32 %0, %1, off scope:SCOPE_DEV"
               :: "v"(p), "v"(v) : "memory");
}

__global__ __launch_bounds__(256) void gcn_spmm_atomic(
    const int* __restrict__ rows, const int* __restrict__ cols,
    const float* __restrict__ vals, const float* __restrict__ S,
    float* __restrict__ out)
{
  const int wave = threadIdx.x >> 5;
  const int lane = threadIdx.x & 31;
  const int edge = blockIdx.x * 8 + wave;
  if (edge >= N_EDGES) return;

  // Edge lists are streamed exactly once -> non-temporal.
  const int   r = __builtin_nontemporal_load(rows + edge);
  const int   c = __builtin_nontemporal_load(cols + edge);
  const float v = __builtin_nontemporal_load(vals + edge);

  const float4 s4 = *(const float4*)(S + (size_t)c * OUT_FEAT + lane * 4);
  float* dst = out + (size_t)r * OUT_FEAT + lane * 4;

  global_fadd(dst + 0, v * s4.x);
  global_fadd(dst + 1, v * s4.y);
  global_fadd(dst + 2, v * s4.z);
  global_fadd(dst + 3, v * s4.w);
}

// ---------------------------------------------------------------------------
extern "C" void kernel_launch(void* const* d_in, const int* in_sizes, int n_in,
                              void* d_out, int out_size, void* d_ws, size_t ws_size,
                              hipStream_t stream) {
  const float* X    = (const float*)d_in[0];   // [100000, 256]
  const int*   erow = (const int*)  d_in[1];   // [1.6M]
  const int*   ecol = (const int*)  d_in[2];   // [1.6M]
  const float* eval = (const float*)d_in[3];   // [1.6M]
  const float* W    = (const float*)d_in[4];   // [256, 128]
  const float* bias = (const float*)d_in[5];   // [128]
  float* out     = (float*)d_out;              // [100000, 128]
  float* support = (float*)d_ws;               // scratch: 100000*128*4 = 51.2 MB

  const int gemmBlocks = (N_NODES + 127) / 128;          // 782
  const int initBlocks = (N_NODES * OUT_FEAT) / 256;     // 50000 (exact)
  const int spmmBlocks = (N_EDGES + 7) / 8;              // 200000

  gcn_gemm_wmma  <<<gemmBlocks, 256, 0, stream>>>(X, W, support);
  gcn_bias_init  <<<initBlocks, 256, 0, stream>>>(bias, out);
  gcn_spmm_atomic<<<spmmBlocks, 256, 0, stream>>>(erow, ecol, eval, support, out);
}